// VectorFieldModule_75307956568293
// MI455X (gfx1250) — compile-verified
//
#include <hip/hip_runtime.h>

// MI455X / gfx1250, wave32. Separable 3D Gaussian blur (25-tap, radius 12)
// as three 1-D passes, each a banded-Toeplitz matmul on the fp32 WMMA pipe
// (V_WMMA_F32_16X16X4_F32). Async global->LDS staging (ASYNCcnt path).

typedef float v2f __attribute__((ext_vector_type(2)));
typedef float v8f __attribute__((ext_vector_type(8)));

#define DIM        160
#define CH_STRIDE  (160*160*160)
#define RADIUS     12
#define TAPS       25
#define SPAN       40              // 16 outputs + 2*RADIUS halo
#define NWAVE      4               // waves per block, one tile per wave

#if __has_builtin(__builtin_amdgcn_global_load_async_to_lds_b32)
#define HAVE_ASYNC_LDS 1
typedef __attribute__((address_space(1))) int as1_int;
typedef __attribute__((address_space(3))) int as3_int;
#else
#define HAVE_ASYNC_LDS 0
#endif

// ---------------------------------------------------------------------------
// g3 = g1 (x) g1 (x) g1 with sum(g1)==1  =>  summing over two axes yields g1.
// ---------------------------------------------------------------------------
__global__ __launch_bounds__(32) void make_g1(const float* __restrict__ kern,
                                              float* __restrict__ g1) {
  int i = threadIdx.x;
  if (i < TAPS) {
    const float* p = kern + i * (TAPS * TAPS);
    float s = 0.0f;
    for (int t = 0; t < TAPS * TAPS; ++t) s += p[t];
    g1[i] = s;
  }
}

// ---------------------------------------------------------------------------
// One separable pass. Each wave computes a 16(conv) x 16(n) tile:
//   out[m,n] = sum_tap g1[tap] * in[x0+m+tap-12, n]
// via 10 accumulating V_WMMA_F32_16X16X4_F32 over the 40-wide band:
//   A_j[m,k] = g1[j+k-m]   (invariant -> registers, from zero-padded LDS)
//   B_j[k,n] = tile[j+k][n]
// Lane mapping (32-bit operands, wave32):
//   A: m=lane&15, VGPR0 -> K=2*(lane>=16), VGPR1 -> K+1   (B mirrors with n)
//   C: VGPR r -> M = r + 8*(lane>=16), N = lane&15
// ---------------------------------------------------------------------------
template <int CONV_STRIDE, int N_STRIDE, int OUTER2_STRIDE, bool CONV_FASTEST>
__global__ __launch_bounds__(NWAVE * 32) void blur_pass(
    const float* __restrict__ in, float* __restrict__ out,
    const float* __restrict__ g1) {
  __shared__ float tile [NWAVE][SPAN][17];   // +1 pad vs bank conflicts
  __shared__ float otile[NWAVE][16][17];
  __shared__ float g1pad[NWAVE][64];         // zero-padded, g1 at [16..40]

  const int lane = threadIdx.x & 31;
  const int w    = threadIdx.x >> 5;

  const int m  = lane & 15;
  const int kb = (lane >> 4) << 1;           // 0 | 2
  const int n  = lane & 15;

  // ---- prologue: zero-padded g1, then hoist A into registers (per wave) ----
  g1pad[w][lane]      = 0.0f;
  g1pad[w][lane + 32] = 0.0f;
  if (lane < TAPS) g1pad[w][16 + lane] = g1[lane];   // LDS in-order per wave

  const int tileIdx = (int)blockIdx.x * NWAVE + w;   // 0..99
  const int tconv   = tileIdx / 10;
  const int tn      = tileIdx - tconv * 10;
  const int x0      = tconv * 16;
  const int n0      = tn * 16;
  const int base    = (int)blockIdx.z * CH_STRIDE + (int)blockIdx.y * OUTER2_STRIDE;

  v2f a_arr[10];
  {
    const int abase = 16 + kb - m;                   // always >= 1 after +j
    asm volatile("s_wait_dscnt 0" ::: "memory");
    #pragma unroll
    for (int c = 0; c < 10; ++c) {
      a_arr[c].x = g1pad[w][abase + c * 4];
      a_arr[c].y = g1pad[w][abase + c * 4 + 1];
    }
  }

  // ---- stage 40x16 input patch into LDS (zero-padded along conv axis) ----
  #pragma unroll
  for (int i = 0; i < SPAN * 16 / 32; ++i) {
    int e = i * 32 + lane;
    int cx, nn;
    if (CONV_FASTEST) { cx = e % SPAN; nn = e / SPAN; }  // conv stride==1
    else              { nn = e & 15;   cx = e >> 4;   }  // n stride==1
    int gx = x0 - RADIUS + cx;
    const float* gp = in + (base + gx * CONV_STRIDE + (n0 + nn) * N_STRIDE);
#if HAVE_ASYNC_LDS
    if ((unsigned)gx < (unsigned)DIM) {
      __builtin_amdgcn_global_load_async_to_lds_b32(
          (as1_int*)(int*)const_cast<float*>(gp),
          (as3_int*)(int*)&tile[w][cx][nn],
          /*offset=*/0, /*cpol=*/0);
    } else {
      tile[w][cx][nn] = 0.0f;
    }
#else
    float v = 0.0f;
    if ((unsigned)gx < (unsigned)DIM) v = *gp;
    tile[w][cx][nn] = v;
#endif
  }
#if HAVE_ASYNC_LDS
  asm volatile("s_wait_asynccnt 0" ::: "memory");
#endif
  asm volatile("s_wait_dscnt 0" ::: "memory");

  // ---- load all B slices, then back-to-back accumulating WMMAs ----
  v2f b_arr[10];
  #pragma unroll
  for (int c = 0; c < 10; ++c) {
    b_arr[c].x = tile[w][c * 4 + kb    ][n];
    b_arr[c].y = tile[w][c * 4 + kb + 1][n];
  }
  asm volatile("s_wait_dscnt 0" ::: "memory");

  v8f acc = {};
  #pragma unroll
  for (int c = 0; c < 10; ++c) {
    acc = __builtin_amdgcn_wmma_f32_16x16x4_f32(
        false, a_arr[c], false, b_arr[c], (short)0, acc, false, false);
  }

  // ---- stage result through LDS, then coalesced global stores ----
  #pragma unroll
  for (int r = 0; r < 8; ++r) {
    int mm = r + ((lane >> 4) << 3);
    otile[w][mm][n] = acc[r];
  }
  asm volatile("s_wait_dscnt 0" ::: "memory");
  #pragma unroll
  for (int i = 0; i < 8; ++i) {
    int e = i * 32 + lane;
    int mm, nn;
    if (CONV_FASTEST) { mm = e & 15; nn = e >> 4; }
    else              { nn = e & 15; mm = e >> 4; }
    out[base + (x0 + mm) * CONV_STRIDE + (n0 + nn) * N_STRIDE] = otile[w][mm][nn];
  }
}

// ---------------------------------------------------------------------------
// Pipeline: x --W--> out --H--> ws --D--> out.  ws: 1KB header + 49.2MB volume.
// ---------------------------------------------------------------------------
extern "C" void kernel_launch(void* const* d_in, const int* in_sizes, int n_in,
                              void* d_out, int out_size, void* d_ws, size_t ws_size,
                              hipStream_t stream) {
  const float* x    = (const float*)d_in[0];
  const float* kern = (const float*)d_in[1];
  float* out = (float*)d_out;
  float* g1  = (float*)d_ws;
  float* tmp = (float*)d_ws + 256;

  make_g1<<<dim3(1), dim3(32), 0, stream>>>(kern, g1);

  dim3 grid(100 / NWAVE, DIM, 3);
  dim3 blk(NWAVE * 32);

  // Pass 1: conv along W (stride 1);   n = H (stride 160); outer2 = D
  blur_pass<1, DIM, DIM * DIM, true><<<grid, blk, 0, stream>>>(x, out, g1);
  // Pass 2: conv along H (stride 160); n = W (stride 1);   outer2 = D
  blur_pass<DIM, 1, DIM * DIM, false><<<grid, blk, 0, stream>>>(out, tmp, g1);
  // Pass 3: conv along D (stride 160*160); n = W (stride 1); outer2 = H
  blur_pass<DIM * DIM, 1, DIM, false><<<grid, blk, 0, stream>>>(tmp, out, g1);
}